// CenterLoss_19653770346899
// MI455X (gfx1250) — compile-verified
//
#include <hip/hip_runtime.h>
#include <math.h>

// Problem constants (match reference)
#define BB   1024
#define NN   128
#define FEAT 512
#define HID  256

typedef float v2f __attribute__((ext_vector_type(2)));
typedef float v8f __attribute__((ext_vector_type(8)));

// ---------------------------------------------------------------------------
// Kernel A: x_proj = x @ W_proj + b_proj   (1024x512 @ 512x256)
// One wave computes one 16x16 tile of x_proj via chained f32 WMMA.
// A-frag layout (16x4 f32): lane = m + 16*(k>>1), vgpr = k&1
//   -> lane reads 2 consecutive K values => one float2 load.
// ---------------------------------------------------------------------------
__global__ __launch_bounds__(32) void proj_kernel(const float* __restrict__ x,
                                                  const float* __restrict__ W,
                                                  const float* __restrict__ bias,
                                                  float* __restrict__ xproj) {
    const int bt   = blockIdx.x;          // 0..63  (batch tile of 16)
    const int ct   = blockIdx.y;          // 0..15  (col tile of 16)
    const int lane = threadIdx.x;         // wave32
    const int m16  = lane & 15;
    const int half = lane >> 4;

    v8f c = {};
    const float* aBase = x + (size_t)(bt * 16 + m16) * FEAT + 2 * half;
    const float* bBase = W + (size_t)(2 * half) * HID + ct * 16 + m16;

    for (int kt = 0; kt < FEAT / 4; ++kt) {
        v2f a = *(const v2f*)(aBase + 4 * kt);          // A[m][k], A[m][k+1]
        v2f b;
        b.x = bBase[(size_t)(4 * kt) * HID];            // B[k][n]
        b.y = bBase[(size_t)(4 * kt) * HID + HID];      // B[k+1][n]
        c = __builtin_amdgcn_wmma_f32_16x16x4_f32(false, a, false, b,
                                                  (short)0, c, false, false);
    }

    const float bv = bias[ct * 16 + m16];
#pragma unroll
    for (int r = 0; r < 8; ++r) {
        // C layout: vgpr r, lanes 0-15 -> row r, lanes 16-31 -> row r+8
        xproj[(size_t)(bt * 16 + r + 8 * half) * HID + ct * 16 + m16] = c[r] + bv;
    }
}

// ---------------------------------------------------------------------------
// Kernel B: dist[b,n] = sqrt(delta^T M_n delta + 1e-12)
// Workgroup = ONE wave32, handles 32 batch rows x 1 center:
//  - two 16-row A tiles share every B fragment (2 WMMAs per B load)
//  - sDelta: 32x256 delta tile, row stride 260 (pad) -> conflict-free b64 A frags
//  - sPanel: 256x16 column-panel of M_n, row-major, stride 20 floats so each
//    16-B chunk is 16-B aligned for GLOBAL_LOAD_ASYNC_TO_LDS_B128 and the two
//    B-frag ds_load_b32 reads are bank-conflict free
//  - software pipeline: issue async copies for panel jj, overlap with the
//    elementwise (Y .* Delta) stage of panel jj-1, then s_wait_asynccnt 0
//    and run the 64-step WMMA chain. Single buffer is safe: panel jj-1 reads
//    were consumed before the new copies are issued (same-wave ordering).
// ---------------------------------------------------------------------------
__global__ __launch_bounds__(32) void mahal_kernel(const float* __restrict__ xproj,
                                                   const float* __restrict__ means,
                                                   const float* __restrict__ invcov,
                                                   float* __restrict__ distmat) {
    __shared__ __attribute__((aligned(16))) float sDelta[32 * 260];
    __shared__ __attribute__((aligned(16))) float sPanel[256 * 20];

    const int lane = threadIdx.x;         // single wave32
    const int n    = blockIdx.y;          // center index 0..127
    const int b0   = blockIdx.x * 32;     // batch tile base
    const int m16  = lane & 15;
    const int half = lane >> 4;

    const float* Mn = invcov + (size_t)n * HID * HID;

    // ---- stage Delta = x_proj[b0..b0+31,:] - means[n,:] (float4 vectorized)
    for (int i = lane; i < 32 * (HID / 4); i += 32) {
        const int rr = i >> 6;            // row 0..31
        const int h4 = (i & 63) << 2;     // col 0..252 step 4
        float4 xv = *(const float4*)(xproj + (size_t)(b0 + rr) * HID + h4);
        float4 mv = *(const float4*)(means + (size_t)n * HID + h4);
        float4 dv = make_float4(xv.x - mv.x, xv.y - mv.y, xv.z - mv.z, xv.w - mv.w);
        *(float4*)(&sDelta[rr * 260 + h4]) = dv;
    }

    float acc0[8], acc1[8];
#pragma unroll
    for (int r = 0; r < 8; ++r) { acc0[r] = 0.0f; acc1[r] = 0.0f; }

    const float* aBase0 = &sDelta[(0  + m16) * 260 + 2 * half];
    const float* aBase1 = &sDelta[(16 + m16) * 260 + 2 * half];

    v8f cP0 = {}, cP1 = {};               // Y tiles of previous panel

    for (int jj = 0; jj < 16; ++jj) {
        // ---- (1) issue async DMA of M_n[:, jj*16 .. +16) into sPanel
        for (int i = lane; i < 256 * 4; i += 32) {
            const int k  = i >> 2;
            const int q4 = (i & 3) << 2;
            unsigned     laddr = (unsigned)(size_t)&sPanel[k * 20 + q4];
            const float* gaddr = Mn + (size_t)k * HID + jj * 16 + q4;
            asm volatile("global_load_async_to_lds_b128 %0, %1, off"
                         :: "v"(laddr), "v"(gaddr) : "memory");
        }
        // warm the next panel in L2 while we are at it
        if (jj + 1 < 16)
            __builtin_prefetch(Mn + (size_t)(lane * 8) * HID + (jj + 1) * 16, 0, 0);

        // ---- (2) overlap: elementwise (Y .* Delta) for panel jj-1
        if (jj > 0) {
            const int col = (jj - 1) * 16 + m16;
#pragma unroll
            for (int r = 0; r < 8; ++r) {
                acc0[r] += cP0[r] * sDelta[(r + 8 * half) * 260 + col];
                acc1[r] += cP1[r] * sDelta[(16 + r + 8 * half) * 260 + col];
            }
        }

        // ---- (3) wait for the DMA, then the WMMA chain for panel jj
        asm volatile("s_wait_asynccnt 0x0" ::: "memory");

        v8f c0 = {}, c1 = {};
        for (int kt = 0; kt < 64; ++kt) {
            const int krow = 4 * kt + 2 * half;
            v2f b;
            b.x = sPanel[krow * 20 + m16];        // B[k  ][n]
            b.y = sPanel[(krow + 1) * 20 + m16];  // B[k+1][n]
            v2f a0 = *(const v2f*)(aBase0 + 4 * kt);
            v2f a1 = *(const v2f*)(aBase1 + 4 * kt);
            c0 = __builtin_amdgcn_wmma_f32_16x16x4_f32(false, a0, false, b,
                                                       (short)0, c0, false, false);
            c1 = __builtin_amdgcn_wmma_f32_16x16x4_f32(false, a1, false, b,
                                                       (short)0, c1, false, false);
        }
        cP0 = c0; cP1 = c1;
    }

    // ---- tail elementwise stage for panel 15
    {
        const int col = 15 * 16 + m16;
#pragma unroll
        for (int r = 0; r < 8; ++r) {
            acc0[r] += cP0[r] * sDelta[(r + 8 * half) * 260 + col];
            acc1[r] += cP1[r] * sDelta[(16 + r + 8 * half) * 260 + col];
        }
    }

    // ---- row sums: reduce across the 16 lanes of each half-wave
#pragma unroll
    for (int r = 0; r < 8; ++r) {
        float v0 = acc0[r], v1 = acc1[r];
        v0 += __shfl_xor(v0, 1); v1 += __shfl_xor(v1, 1);
        v0 += __shfl_xor(v0, 2); v1 += __shfl_xor(v1, 2);
        v0 += __shfl_xor(v0, 4); v1 += __shfl_xor(v1, 4);
        v0 += __shfl_xor(v0, 8); v1 += __shfl_xor(v1, 8);
        if (m16 == 0) {
            const int row = r + 8 * half;
            distmat[(size_t)(b0 + row) * NN + n]      = sqrtf(v0 + 1e-12f);
            distmat[(size_t)(b0 + 16 + row) * NN + n] = sqrtf(v1 + 1e-12f);
        }
    }
}

// ---------------------------------------------------------------------------
// Kernel C: loss = sum_b clip(distmat[b, labels[b]], 1e-12, 1e12) / B
// ---------------------------------------------------------------------------
__global__ __launch_bounds__(256) void loss_kernel(const int* __restrict__ labels,
                                                   const float* __restrict__ distmat,
                                                   float* __restrict__ out) {
    __shared__ float red[256];
    const int t = threadIdx.x;
    float s = 0.0f;
    for (int b = t; b < BB; b += 256) {
        const int lab = labels[b];
        float d = distmat[(size_t)b * NN + lab];
        d = fminf(fmaxf(d, 1e-12f), 1e12f);
        s += d;
    }
    red[t] = s;
    __syncthreads();
    for (int off = 128; off > 0; off >>= 1) {
        if (t < off) red[t] += red[t + off];
        __syncthreads();
    }
    if (t == 0) out[0] = red[0] / (float)BB;
}

// ---------------------------------------------------------------------------
extern "C" void kernel_launch(void* const* d_in, const int* in_sizes, int n_in,
                              void* d_out, int out_size, void* d_ws, size_t ws_size,
                              hipStream_t stream) {
    const float* x      = (const float*)d_in[0];
    const int*   labels = (const int*)d_in[1];
    const float* W      = (const float*)d_in[2];
    const float* bproj  = (const float*)d_in[3];
    const float* means  = (const float*)d_in[4];
    const float* invcov = (const float*)d_in[5];

    float* out   = (float*)d_out;      // out[0] = loss, out[1..] = distmat (1024x128)
    float* xproj = (float*)d_ws;       // 1024x256 f32 = 1 MB scratch

    proj_kernel<<<dim3(BB / 16, HID / 16), 32, 0, stream>>>(x, W, bproj, xproj);
    mahal_kernel<<<dim3(BB / 32, NN), 32, 0, stream>>>(xproj, means, invcov, out + 1);
    loss_kernel<<<1, 256, 0, stream>>>(labels, out + 1, out);
}